// BilateralGrid_87943750353522
// MI455X (gfx1250) — compile-verified
//
#include <hip/hip_runtime.h>
#include <cstdint>
#include <cstddef>

#define TPB 256
#define GRID_PER_BATCH 24576   // 12 ch * 16 d * 16 y * 8 x floats (96 KB)
#define PIX_PER_BATCH (1 << 20)

typedef __attribute__((address_space(1))) int gint_t;   // global (AS1)
typedef __attribute__((address_space(3))) int lint_t;   // LDS (AS3)

// Accumulate 12 channels of one grid corner (channel-last, 48B record) into c[].
__device__ __forceinline__ void acc12(const float* __restrict__ s, int off,
                                      float wk, float c[12]) {
  const float4* p = reinterpret_cast<const float4*>(s + off);  // 16B aligned (off % 12 == 0 -> 48B)
  float4 a = p[0], b = p[1], d = p[2];                          // 3x ds_load_b128
  c[0]  = fmaf(wk, a.x, c[0]);   c[1]  = fmaf(wk, a.y, c[1]);
  c[2]  = fmaf(wk, a.z, c[2]);   c[3]  = fmaf(wk, a.w, c[3]);
  c[4]  = fmaf(wk, b.x, c[4]);   c[5]  = fmaf(wk, b.y, c[5]);
  c[6]  = fmaf(wk, b.z, c[6]);   c[7]  = fmaf(wk, b.w, c[7]);
  c[8]  = fmaf(wk, d.x, c[8]);   c[9]  = fmaf(wk, d.y, c[9]);
  c[10] = fmaf(wk, d.z, c[10]);  c[11] = fmaf(wk, d.w, c[11]);
}

__global__ __launch_bounds__(TPB) void bilateral_grid_apply(
    const float* __restrict__ grid,   // (B, 12, 16, 16, 8)
    const float* __restrict__ image,  // (B, 3, 1024, 1024)
    float* __restrict__ out)          // (B, 3, 1024, 1024)
{
  extern __shared__ float sgrid[];    // (d, y, x, c) channel-last, 96 KB
  const int b   = blockIdx.y;
  const int tid = threadIdx.x;
  const float* gbase = grid + (size_t)b * GRID_PER_BATCH;

  // ---- Stage this batch's grid into LDS with an on-the-fly transpose.
  // Global side: consecutive lanes read consecutive floats (coalesced b32).
  // LDS side: per-lane destination address scatters into channel-last layout.
  // CDNA5 async global->LDS DMA, tracked by ASYNCcnt.
  for (int i = tid; i < GRID_PER_BATCH; i += TPB) {
    const int c    = i >> 11;        // channel 0..11 (channel stride = 2048 floats)
    const int rest = i & 2047;       // (d*16 + y)*8 + x
    const float* src = gbase + i;    // == gbase + c*2048 + rest
    float* dst = &sgrid[rest * 12 + c];
#if __has_builtin(__builtin_amdgcn_global_load_async_to_lds_b32)
    __builtin_amdgcn_global_load_async_to_lds_b32(
        (gint_t*)src, (lint_t*)dst, /*offset=*/0, /*cpol=*/0);
#else
    uint32_t lds_off = (uint32_t)(size_t)dst;   // low 32 bits of generic = LDS byte addr
    asm volatile("global_load_async_to_lds_b32 %0, %1, off"
                 :: "v"(lds_off), "v"(src) : "memory");
#endif
  }
#if __has_builtin(__builtin_amdgcn_s_wait_asynccnt)
  __builtin_amdgcn_s_wait_asynccnt(0);
#else
  asm volatile("s_wait_asynccnt 0" ::: "memory");
#endif
  __syncthreads();

  const float* imR = image + ((size_t)b * 3 + 0) * PIX_PER_BATCH;
  const float* imG = image + ((size_t)b * 3 + 1) * PIX_PER_BATCH;
  const float* imB = image + ((size_t)b * 3 + 2) * PIX_PER_BATCH;
  float* oR = out + ((size_t)b * 3 + 0) * PIX_PER_BATCH;
  float* oG = out + ((size_t)b * 3 + 1) * PIX_PER_BATCH;
  float* oB = out + ((size_t)b * 3 + 2) * PIX_PER_BATCH;

  // 4 pixels per thread per iteration, b128 vectorized image I/O.
  // 1M pixels / (gridDim.x * TPB * 4) = 8 full iterations when gridDim.x = 128.
  const int stride4 = gridDim.x * TPB * 4;
  for (int idx4 = (blockIdx.x * TPB + tid) * 4; idx4 < PIX_PER_BATCH;
       idx4 += stride4) {
    const float4 r4 = *reinterpret_cast<const float4*>(imR + idx4);
    const float4 g4 = *reinterpret_cast<const float4*>(imG + idx4);
    const float4 b4 = *reinterpret_cast<const float4*>(imB + idx4);
    const float rr[4] = {r4.x, r4.y, r4.z, r4.w};
    const float gg[4] = {g4.x, g4.y, g4.z, g4.w};
    const float bb[4] = {b4.x, b4.y, b4.z, b4.w};
    float o0[4], o1[4], o2[4];

    // All 4 pixels share a row: h constant across the quad.
    const int h  = idx4 >> 10;
    const int w0 = idx4 & 1023;
    const float uy = fminf(h * (15.0f / 1023.0f), 15.0f);
    const float fy = floorf(uy);
    const int y0 = (int)fy;
    const int y1 = min(y0 + 1, 15);
    const float wy = uy - fy, vy = 1.0f - wy;
    const int ry0 = y0 << 3, ry1 = y1 << 3;

#pragma unroll
    for (int j = 0; j < 4; ++j) {
      const float r  = rr[j];
      const float g  = gg[j];
      const float bl = bb[j];

      // Guide luminance -> z coordinate (align_corners=True, border clamp).
      const float guide = fmaf(0.299f, r, fmaf(0.587f, g, 0.114f * bl));
      float ux = fminf((w0 + j) * (7.0f / 1023.0f), 7.0f);
      float uz = fminf(fmaxf(guide * 15.0f, 0.0f), 15.0f);

      const float fx = floorf(ux), fz = floorf(uz);
      const int x0 = (int)fx, z0 = (int)fz;
      const int x1 = min(x0 + 1, 7);
      const int z1 = min(z0 + 1, 15);
      const float wx = ux - fx, wz = uz - fz;
      const float vx = 1.0f - wx, vz = 1.0f - wz;

      // LDS corner offsets (floats): ((d*16 + y)*8 + x) * 12
      const int pz0 = z0 << 7, pz1 = z1 << 7;
      const int o00 = (pz0 + ry0) * 12, o01 = (pz0 + ry1) * 12;
      const int o10 = (pz1 + ry0) * 12, o11 = (pz1 + ry1) * 12;
      const int xo0 = x0 * 12, xo1 = x1 * 12;

      const float wzy00 = vz * vy, wzy01 = vz * wy;
      const float wzy10 = wz * vy, wzy11 = wz * wy;

      float c[12];
#pragma unroll
      for (int k = 0; k < 12; ++k) c[k] = 0.0f;
      acc12(sgrid, o00 + xo0, wzy00 * vx, c);
      acc12(sgrid, o00 + xo1, wzy00 * wx, c);
      acc12(sgrid, o01 + xo0, wzy01 * vx, c);
      acc12(sgrid, o01 + xo1, wzy01 * wx, c);
      acc12(sgrid, o10 + xo0, wzy10 * vx, c);
      acc12(sgrid, o10 + xo1, wzy10 * wx, c);
      acc12(sgrid, o11 + xo0, wzy11 * vx, c);
      acc12(sgrid, o11 + xo1, wzy11 * wx, c);

      // A (3x3, row-major in c[0..8]) * rgb + bias (c[9..11]), clamp to [0,1].
      const float t0 = fmaf(c[0], r, fmaf(c[1], g, fmaf(c[2], bl, c[9])));
      const float t1 = fmaf(c[3], r, fmaf(c[4], g, fmaf(c[5], bl, c[10])));
      const float t2 = fmaf(c[6], r, fmaf(c[7], g, fmaf(c[8], bl, c[11])));
      o0[j] = fminf(fmaxf(t0, 0.0f), 1.0f);
      o1[j] = fminf(fmaxf(t1, 0.0f), 1.0f);
      o2[j] = fminf(fmaxf(t2, 0.0f), 1.0f);
    }

    *reinterpret_cast<float4*>(oR + idx4) = make_float4(o0[0], o0[1], o0[2], o0[3]);
    *reinterpret_cast<float4*>(oG + idx4) = make_float4(o1[0], o1[1], o1[2], o1[3]);
    *reinterpret_cast<float4*>(oB + idx4) = make_float4(o2[0], o2[1], o2[2], o2[3]);
  }
}

extern "C" void kernel_launch(void* const* d_in, const int* in_sizes, int n_in,
                              void* d_out, int out_size, void* d_ws, size_t ws_size,
                              hipStream_t stream) {
  (void)n_in; (void)out_size; (void)d_ws; (void)ws_size;
  const float* grid  = (const float*)d_in[0];   // (B,12,16,16,8) f32
  const float* image = (const float*)d_in[1];   // (B,3,1024,1024) f32
  float* out = (float*)d_out;                   // (B,3,1024,1024) f32

  const int B = in_sizes[0] / GRID_PER_BATCH;   // 4
  // 128 blocks per batch * 8 waves * 96KB LDS -> 3 resident blocks/WGP,
  // each thread streams 8 quads (32 pixels) of b128-coalesced loads/stores.
  dim3 grd(128, (unsigned)B);
  dim3 blk(TPB);
  bilateral_grid_apply<<<grd, blk, GRID_PER_BATCH * sizeof(float), stream>>>(
      grid, image, out);
}